// GCN_78151224918828
// MI455X (gfx1250) — compile-verified
//
#include <hip/hip_runtime.h>
#include <cstddef>

typedef __attribute__((ext_vector_type(2))) float v2f;
typedef __attribute__((ext_vector_type(8))) float v8f;

// ---------------- utility kernels ----------------

__global__ void fill_f32(float* __restrict__ p, float v, size_t n) {
  size_t i = (size_t)blockIdx.x * blockDim.x + threadIdx.x;
  size_t stride = (size_t)gridDim.x * blockDim.x;
  for (; i < n; i += stride) p[i] = v;
}

__global__ void deg_accum(const int* __restrict__ dst, float* __restrict__ deg,
                          long long E) {
  long long e = (long long)blockIdx.x * blockDim.x + threadIdx.x;
  if (e < E) atomicAdd(&deg[dst[e]], 1.0f);
}

__global__ void rsqrt_inplace(float* __restrict__ p, int n) {
  int i = blockIdx.x * blockDim.x + threadIdx.x;
  if (i < n) p[i] = rsqrtf(p[i]);
}

// ---------------- WMMA GEMM: H[N,64] = X[N,K] @ W[K,64] ----------------
// Block: 256 threads = 8 waves; block tile = 32 rows x 64 cols;
// each wave computes one 16x16 tile via V_WMMA_F32_16X16X4_F32.
//
// fp32 WMMA VGPR layouts (CDNA5 ISA):
//  A (16x4): lanes 0-15 hold M=lane, {v0,v1} = K={0,1}; lanes 16-31 K={2,3}
//  B (4x16): lanes 0-15 hold N=lane, {v0,v1} = K={0,1}; lanes 16-31 K={2,3}
//  C/D (16x16): v[i], lanes 0-15 -> M=i, N=lane; lanes 16-31 -> M=i+8, N=lane-16

template <int K>
__global__ __launch_bounds__(256) void gemm_wmma(const float* __restrict__ X,
                                                 const float* __restrict__ W,
                                                 float* __restrict__ H, int n) {
  __shared__ float sW[K * 64];   // K x 64
  __shared__ float sX[32 * K];   // 32 x K
  const int tid  = threadIdx.x;
  const int row0 = blockIdx.x * 32;

  for (int i = tid; i < K * 64; i += 256) sW[i] = W[i];
  for (int i = tid; i < 32 * K; i += 256) {
    int r = row0 + i / K;
    if (r > n - 1) r = n - 1;           // clamp (keeps EXEC all-ones later)
    sX[i] = X[(size_t)r * K + (i % K)];
  }
  __syncthreads();

  const int lane = tid & 31;
  const int wave = tid >> 5;
  const int wrow = (wave >> 2) << 4;    // 0 or 16
  const int wcol = (wave & 3) << 4;     // 0,16,32,48
  const int l15  = lane & 15;
  const int lhi  = lane >> 4;           // 0 or 1

  v8f acc = {};
#pragma unroll 8
  for (int k0 = 0; k0 < K; k0 += 4) {
    const int ka = k0 + (lhi << 1);     // this half-wave's K pair
    v2f a, b;
    a.x = sX[(wrow + l15) * K + ka];
    a.y = sX[(wrow + l15) * K + ka + 1];
    b.x = sW[ka * 64 + wcol + l15];
    b.y = sW[(ka + 1) * 64 + wcol + l15];
    acc = __builtin_amdgcn_wmma_f32_16x16x4_f32(
        /*neg_a=*/false, a, /*neg_b=*/false, b,
        /*c_mod=*/(short)0, acc, /*reuse_a=*/false, /*reuse_b=*/false);
  }

  const int orow = row0 + wrow + (lhi << 3);
  const int ocol = wcol + l15;
#pragma unroll
  for (int i = 0; i < 8; ++i) {
    const int r = orow + i;
    if (r < n) H[(size_t)r * 64 + ocol] = acc[i];
  }
}

// ---------------- edge aggregation: AGG[dst] += H[src] * norm ----------------
// 16 threads per edge, float4 (B128) gather per thread, 4 f32 atomics.

__global__ void aggregate(const float* __restrict__ H, const int* __restrict__ src,
                          const int* __restrict__ dst, const float* __restrict__ dis,
                          float* __restrict__ AGG, long long E) {
  long long t = (long long)blockIdx.x * blockDim.x + threadIdx.x;
  long long e = t >> 4;
  if (e >= E) return;
  const int cg = ((int)t & 15) << 2;    // channel group base (0..60)
  const int s = src[e];
  const int d = dst[e];
  const float norm = dis[s] * dis[d];
  const float4 hv = *reinterpret_cast<const float4*>(H + (size_t)s * 64 + cg);
  float* o = AGG + (size_t)d * 64 + cg;
  atomicAdd(o + 0, hv.x * norm);
  atomicAdd(o + 1, hv.y * norm);
  atomicAdd(o + 2, hv.z * norm);
  atomicAdd(o + 3, hv.w * norm);
}

// ---------------- epilogue: out = agg + h*dis^2 + bias, optional ReLU --------

__global__ void finalize(const float* __restrict__ H, const float* __restrict__ AGG,
                         const float* __restrict__ dis, const float* __restrict__ bias,
                         float* __restrict__ OUT, int relu, long long total) {
  long long idx = (long long)blockIdx.x * blockDim.x + threadIdx.x;
  if (idx >= total) return;
  const int i = (int)(idx >> 6);
  const int c = (int)(idx & 63);
  const float di = dis[i];
  float v = AGG[idx] + H[idx] * di * di + bias[c];
  if (relu) v = fmaxf(v, 0.0f);
  OUT[idx] = v;
}

// ---------------- launcher ----------------

extern "C" void kernel_launch(void* const* d_in, const int* in_sizes, int n_in,
                              void* d_out, int out_size, void* d_ws, size_t ws_size,
                              hipStream_t stream) {
  const float* x  = (const float*)d_in[0];
  const int*   ei = (const int*)d_in[1];     // edge_index [2, E] (int32)
  const float* W1 = (const float*)d_in[2];
  const float* b1 = (const float*)d_in[3];
  const float* W2 = (const float*)d_in[4];
  const float* b2 = (const float*)d_in[5];
  const float* W3 = (const float*)d_in[6];
  const float* b3 = (const float*)d_in[7];
  float* out = (float*)d_out;

  const int N = in_sizes[0] / 128;
  const long long E = (long long)in_sizes[1] / 2;
  const int* src = ei;
  const int* dst = ei + E;

  float* dis  = (float*)d_ws;                // [N]
  float* bufA = dis + N;                     // [N*64]  (h)
  float* bufB = bufA + (size_t)N * 64;       // [N*64]  (agg / layer output)

  const long long NC = (long long)N * 64;
  dim3 blk(256);
  auto cdiv = [](long long a, long long b) { return (unsigned)((a + b - 1) / b); };

  // degrees: deg = 1 (self-loop) + incoming edge count; dis = rsqrt(deg)
  fill_f32<<<1024, blk, 0, stream>>>(dis, 1.0f, (size_t)N);
  deg_accum<<<cdiv(E, 256), blk, 0, stream>>>(dst, dis, E);
  rsqrt_inplace<<<cdiv(N, 256), blk, 0, stream>>>(dis, N);

  const unsigned gemm_blocks = cdiv(N, 32);
  const unsigned agg_blocks  = cdiv(E * 16, 256);
  const unsigned fin_blocks  = cdiv(NC, 256);

  // Layer 1: x[N,128] @ W1 -> bufA; agg -> bufB; epilogue in-place (ReLU)
  gemm_wmma<128><<<gemm_blocks, blk, 0, stream>>>(x, W1, bufA, N);
  fill_f32<<<2048, blk, 0, stream>>>(bufB, 0.0f, (size_t)NC);
  aggregate<<<agg_blocks, blk, 0, stream>>>(bufA, src, dst, dis, bufB, E);
  finalize<<<fin_blocks, blk, 0, stream>>>(bufA, bufB, dis, b1, bufB, 1, NC);

  // Layer 2: bufB @ W2 -> bufA; agg -> d_out; epilogue in-place (ReLU)
  gemm_wmma<64><<<gemm_blocks, blk, 0, stream>>>(bufB, W2, bufA, N);
  fill_f32<<<2048, blk, 0, stream>>>(out, 0.0f, (size_t)NC);
  aggregate<<<agg_blocks, blk, 0, stream>>>(bufA, src, dst, dis, out, E);
  finalize<<<fin_blocks, blk, 0, stream>>>(bufA, out, dis, b2, out, 1, NC);

  // Layer 3: d_out @ W3 -> bufA; agg -> bufB; epilogue -> d_out (no ReLU)
  gemm_wmma<64><<<gemm_blocks, blk, 0, stream>>>(out, W3, bufA, N);
  fill_f32<<<2048, blk, 0, stream>>>(bufB, 0.0f, (size_t)NC);
  aggregate<<<agg_blocks, blk, 0, stream>>>(bufA, src, dst, dis, bufB, E);
  finalize<<<fin_blocks, blk, 0, stream>>>(bufA, bufB, dis, b3, out, 0, NC);
}